// GCNLayer_27599459844749
// MI455X (gfx1250) — compile-verified
//
#include <hip/hip_runtime.h>
#include <hip/hip_bf16.h>

// GCN layer on MI455X (gfx1250), sparse formulation.
// N=8192 nodes, E=262144 edges, IN_F=OUT_F=128.

#define NNODES 8192
#define NEDGES 262144
#define NFEAT  128

typedef __attribute__((ext_vector_type(2))) float v2f;
typedef __attribute__((ext_vector_type(8))) float v8f;

// ---------------------------------------------------------------------------
// Kernel 1: support = X @ W via V_WMMA_F32_16X16X4_F32.
// Block = 256 threads = 8 waves. Each block owns one 16-row M tile; each wave
// owns one 16-col N tile (8 waves x 16 = 128 output features).
// f32 WMMA lane layouts (ISA 7.12.2):
//   A (16x4): lanes 0-15 -> M=lane, vgpr{0,1}=K{k0,k0+1};
//             lanes 16-31 -> M=lane-16, vgpr{0,1}=K{k0+2,k0+3}
//   B (4x16): vgpr0 = row K=k0(+2 for hi lanes), vgpr1 = next K row; N=lane%16
//   C (16x16): vgpr r -> M = r (lanes 0-15) / r+8 (lanes 16-31); N = lane%16
// ---------------------------------------------------------------------------
__global__ __launch_bounds__(256) void gcn_support_gemm(
    const float* __restrict__ x,       // [NNODES, NFEAT]
    const float* __restrict__ w,       // [NFEAT, NFEAT]
    float* __restrict__ support)       // [NNODES, NFEAT]
{
    const int wave = threadIdx.x >> 5;       // 0..7 -> N tile
    const int lane = threadIdx.x & 31;
    const int ln   = lane & 15;
    const int hi   = lane >> 4;              // 0 for lanes 0-15, 1 for 16-31
    const int m0   = blockIdx.x << 4;        // M tile base
    const int n0   = wave << 4;              // N tile base

    const float* arow = x + (size_t)(m0 + ln) * NFEAT;

    v8f c = {};
#pragma unroll
    for (int k0 = 0; k0 < NFEAT; k0 += 4) {
        const int ka = k0 + 2 * hi;
        // A fragment: two consecutive K values of this lane's row (8B aligned)
        v2f a = *(const v2f*)(arow + ka);
        // B fragment: two K rows of W at column n0+ln
        v2f b;
        b.x = w[(size_t)(ka    ) * NFEAT + n0 + ln];
        b.y = w[(size_t)(ka + 1) * NFEAT + n0 + ln];
        c = __builtin_amdgcn_wmma_f32_16x16x4_f32(
                /*neg_a=*/false, a, /*neg_b=*/false, b,
                /*c_mod=*/(short)0, c, /*reuse_a=*/false, /*reuse_b=*/false);
    }

#pragma unroll
    for (int r = 0; r < 8; ++r) {
        support[(size_t)(m0 + r + 8 * hi) * NFEAT + n0 + ln] = c[r];
    }
}

// ---------------------------------------------------------------------------
// Kernel 2: deg[i] = 1.0 (self-loop from +I)
// ---------------------------------------------------------------------------
__global__ __launch_bounds__(256) void gcn_deg_init(float* __restrict__ deg)
{
    int i = blockIdx.x * blockDim.x + threadIdx.x;
    if (i < NNODES) deg[i] = 1.0f;
}

// ---------------------------------------------------------------------------
// Kernel 3: deg[row[e]] += edge_w[e]  (duplicates sum, matching to_dense)
// ---------------------------------------------------------------------------
__global__ __launch_bounds__(256) void gcn_deg_accum(
    const long long* __restrict__ adj,   // [2, E] int64: rows then cols
    const float* __restrict__ ew,        // [E]
    float* __restrict__ deg)
{
    int e = blockIdx.x * blockDim.x + threadIdx.x;
    if (e < NEDGES) {
        int r = (int)adj[e];
        atomicAdd(&deg[r], ew[e]);
    }
}

// ---------------------------------------------------------------------------
// Kernel 4: deg -> d_inv_sqrt in place
// ---------------------------------------------------------------------------
__global__ __launch_bounds__(256) void gcn_deg_finalize(float* __restrict__ deg)
{
    int i = blockIdx.x * blockDim.x + threadIdx.x;
    if (i < NNODES) deg[i] = rsqrtf(deg[i] + 1e-10f);
}

// ---------------------------------------------------------------------------
// Kernel 5: out[i,f] = dis[i]^2 * support[i,f] + bias[f]
// (self-loop contribution of norm_adj diagonal + bias; fully initializes out)
// ---------------------------------------------------------------------------
__global__ __launch_bounds__(256) void gcn_out_init(
    const float* __restrict__ support,
    const float* __restrict__ dis,
    const float* __restrict__ bias,
    float* __restrict__ out)
{
    int t = blockIdx.x * blockDim.x + threadIdx.x;   // [0, NNODES*NFEAT)
    int i = t >> 7;          // node
    int f = t & (NFEAT - 1); // feature
    float s = dis[i];
    out[t] = s * s * support[t] + bias[f];
}

// ---------------------------------------------------------------------------
// Kernel 6: per (edge, feature): out[r,f] += dis[r]*w*dis[c] * support[c,f]
// 128 threads per edge; adj/dis/edge_w reads broadcast out of L2/WGP$.
// ---------------------------------------------------------------------------
__global__ __launch_bounds__(256) void gcn_edge_scatter(
    const long long* __restrict__ adj,
    const float* __restrict__ ew,
    const float* __restrict__ dis,
    const float* __restrict__ support,
    float* __restrict__ out)
{
    long long t = (long long)blockIdx.x * blockDim.x + threadIdx.x;
    int e = (int)(t >> 7);
    int f = (int)(t & (NFEAT - 1));
    int r = (int)adj[e];
    int c = (int)adj[NEDGES + e];
    float coef = dis[r] * ew[e] * dis[c];
    atomicAdd(&out[(size_t)r * NFEAT + f], coef * support[(size_t)c * NFEAT + f]);
}

// ---------------------------------------------------------------------------
extern "C" void kernel_launch(void* const* d_in, const int* in_sizes, int n_in,
                              void* d_out, int out_size, void* d_ws, size_t ws_size,
                              hipStream_t stream) {
    const float*     x    = (const float*)d_in[0];      // [8192,128] f32
    const long long* adj  = (const long long*)d_in[1];  // [2,262144] i64
    const float*     ew   = (const float*)d_in[2];      // [262144] f32
    const float*     w    = (const float*)d_in[3];      // [128,128] f32
    const float*     bias = (const float*)d_in[4];      // [128] f32
    float* out = (float*)d_out;                         // [8192,128] f32

    // Workspace layout: support [NNODES*NFEAT] f32, then deg/dis [NNODES] f32.
    float* support = (float*)d_ws;
    float* deg     = support + (size_t)NNODES * NFEAT;

    // 1) support = X @ W  (WMMA f32 16x16x4)
    gcn_support_gemm<<<NNODES / 16, 256, 0, stream>>>(x, w, support);

    // 2) deg = 1 (self-loops)
    gcn_deg_init<<<NNODES / 256, 256, 0, stream>>>(deg);

    // 3) deg[row] += edge_w
    gcn_deg_accum<<<NEDGES / 256, 256, 0, stream>>>(adj, ew, deg);

    // 4) deg -> 1/sqrt(deg+eps)
    gcn_deg_finalize<<<NNODES / 256, 256, 0, stream>>>(deg);

    // 5) out = dis^2 * support + bias
    gcn_out_init<<<(NNODES * NFEAT) / 256, 256, 0, stream>>>(support, deg, bias, out);

    // 6) scatter edges: out[r] += dis[r]*w*dis[c] * support[c]
    gcn_edge_scatter<<<((long long)NEDGES * NFEAT) / 256, 256, 0, stream>>>(
        adj, ew, deg, support, out);
}